// RNN_31585189494799
// MI455X (gfx1250) — compile-verified
//
#include <hip/hip_runtime.h>
#include <cstdint>

// RNN scan: N_IN=2, N_HID=2, N_OUT=1, SEQ=512, BATCH=32768.
// Memory-bound (~192 MB @ 23.3 TB/s ~= 8.2 us floor; working set == 192 MB L2,
// so graph replays run mostly out of L2). One lane per batch element,
// sequential over time. x is staged into LDS via the CDNA5 async global->LDS
// DMA path (ASYNCcnt) in a 32-deep software pipeline: at ~1 wave32/SIMD,
// latency hiding must come from pipeline depth, not occupancy.
// WMMA is deliberately not used: the recurrence applies tanh every step on a
// 2-wide hidden state, so the largest matmul is 2x2 -- two orders below a
// 16x16x32 WMMA tile; packing would only add overhead to a BW-bound kernel.

#define SEQ     512
#define BATCHN  32768
#define BLOCK   128
#define NSTAGE  32          // 32 timesteps in flight; LDS = 32*128*8B = 32 KB

#if __has_builtin(__builtin_amdgcn_tanhf)
  #define FAST_TANH(x) __builtin_amdgcn_tanhf(x)
#elif __has_builtin(__builtin_amdgcn_tanh_f32)
  #define FAST_TANH(x) __builtin_amdgcn_tanh_f32(x)
#else
  #define FAST_TANH(x) tanhf(x)
#endif

// One per-lane 8-byte async copy Global -> LDS (tracked by ASYNCcnt).
// vdst = per-lane LDS byte address, vaddr = per-lane 32-bit byte offset,
// saddr = uniform 64-bit base (SGPR pair).
__device__ __forceinline__ void async_copy_b64(unsigned lds_byte_addr,
                                               unsigned voff_bytes,
                                               const float* saddr) {
  asm volatile("global_load_async_to_lds_b64 %0, %1, %2"
               :: "v"(lds_byte_addr), "v"(voff_bytes), "s"(saddr)
               : "memory");
}

template <int N>
__device__ __forceinline__ void wait_async() {
  asm volatile("s_wait_asynccnt %0" :: "n"(N) : "memory");
}

__global__ __launch_bounds__(BLOCK) void RNN_31585189494799_kernel(
    const float* __restrict__ x,
    const float* __restrict__ W_ih,  const float* __restrict__ W_hh,
    const float* __restrict__ b_ih,  const float* __restrict__ b_hh,
    const float* __restrict__ fc_w,  const float* __restrict__ fc_b,
    float* __restrict__ out)
{
  __shared__ float smem[NSTAGE * BLOCK * 2];

  const int tid = threadIdx.x;
  const int b   = blockIdx.x * BLOCK + tid;

  // Wave-uniform weights (scalar loads).
  const float w00 = W_ih[0], w01 = W_ih[1], w10 = W_ih[2], w11 = W_ih[3];
  const float u00 = W_hh[0], u01 = W_hh[1], u10 = W_hh[2], u11 = W_hh[3];
  const float bb0 = b_ih[0] + b_hh[0];
  const float bb1 = b_ih[1] + b_hh[1];
  const float f0  = fc_w[0], f1 = fc_w[1], fb = fc_b[0];

  // Per-lane global byte offset of x[t][b][0] relative to x + t*BATCH*2.
  const unsigned voff = (unsigned)b * 8u;
  // Wave-relative LDS byte address of this lane's slot in stage 0.
  // (Generic LDS pointer low 32 bits == LDS byte offset on gfx1250.)
  const unsigned lds_lane0 = (unsigned)(uintptr_t)(&smem[tid * 2]);

  // ---- prologue: fill the pipeline with timesteps 0..NSTAGE-1 ----
  #pragma unroll
  for (int s = 0; s < NSTAGE; ++s) {
    async_copy_b64(lds_lane0 + (unsigned)s * (BLOCK * 8u), voff,
                   x + (size_t)s * (BATCHN * 2));
  }

  float h0 = 0.0f, h1 = 0.0f;
  float* outp = out + b;

  #pragma unroll 4
  for (int t = 0; t < SEQ; ++t) {
    // In-order completion + counter <= NSTAGE-1  =>  tile t has landed in LDS.
    wait_async<NSTAGE - 1>();
    const int slot = t & (NSTAGE - 1);
    const float2 xv =
        *reinterpret_cast<const float2*>(&smem[(slot * BLOCK + tid) * 2]);

    // Refill the just-consumed slot. Clamp the source timestep so exactly one
    // load is issued every iteration: this keeps the "<= NSTAGE-1" wait
    // invariant valid through the tail (clamped loads land in slots that are
    // never read again, so their duplicate data is harmless).
    const int tnext = (t + NSTAGE < SEQ) ? (t + NSTAGE) : (SEQ - 1);
    async_copy_b64(lds_lane0 + (unsigned)slot * (BLOCK * 8u), voff,
                   x + (size_t)tnext * (BATCHN * 2));

    // xin = x @ W_ih^T + (b_ih + b_hh); h' = tanh(xin + h @ W_hh^T)
    float a0 = fmaf(w00, xv.x, fmaf(w01, xv.y, bb0));
    float a1 = fmaf(w10, xv.x, fmaf(w11, xv.y, bb1));
    a0 = fmaf(u00, h0, fmaf(u01, h1, a0));
    a1 = fmaf(u10, h0, fmaf(u11, h1, a1));
    h0 = FAST_TANH(a0);
    h1 = FAST_TANH(a1);

    // out[t][b] = exp(h' . fc_w + fc_b)   (streaming store, bypass L2 rinse)
    const float o = __expf(fmaf(f0, h0, fmaf(f1, h1, fb)));
    __builtin_nontemporal_store(o, outp);
    outp += BATCHN;
  }

  // Final hidden state, packed as float2 per lane (coalesced b64 store).
  float2 hv; hv.x = h0; hv.y = h1;
  *reinterpret_cast<float2*>(out + (size_t)SEQ * BATCHN + 2 * (size_t)b) = hv;
}

extern "C" void kernel_launch(void* const* d_in, const int* in_sizes, int n_in,
                              void* d_out, int out_size, void* d_ws, size_t ws_size,
                              hipStream_t stream) {
  const float* x    = (const float*)d_in[0];
  const float* W_ih = (const float*)d_in[1];
  const float* W_hh = (const float*)d_in[2];
  const float* b_ih = (const float*)d_in[3];
  const float* b_hh = (const float*)d_in[4];
  const float* fc_w = (const float*)d_in[5];
  const float* fc_b = (const float*)d_in[6];
  float* out = (float*)d_out;

  dim3 grid(BATCHN / BLOCK);   // 256 blocks x 128 threads = 32768 lanes
  dim3 block(BLOCK);
  RNN_31585189494799_kernel<<<grid, block, 0, stream>>>(
      x, W_ih, W_hh, b_ih, b_hh, fc_w, fc_b, out);
}